// OHEMLoss_2482491097332
// MI455X (gfx1250) — compile-verified
//
#include <hip/hip_runtime.h>
#include <stdint.h>

// ---------------------------------------------------------------------------
// OHEM cross-entropy for logits [B=8,C=19,H=512,W=1024] fp32, labels int32.
// Pipeline:
//   k_init    : zero histograms/scalars (ws is not re-poisoned between replays)
//   k_pass1   : stream logits once (async global->LDS double buffer), online
//               logsumexp, write loss[N], LDS histogram of bits[31:21], n_valid
//   k_scan(0) : compute n_keep, find top-level bucket b1 + rank k1
//   k_histmid : histogram bits[20:10] restricted to b1
//   k_scan(1) : find b2 + rank k2
//   k_histmid : histogram bits[9:0] restricted to (b1,b2)
//   k_scan(2) : exact threshold bits = (b1<<21)|(b2<<10)|b3
//   k_partial : deterministic per-WG sum/count of selected losses
//   k_final   : deterministic tree reduce -> d_out[0]
// ---------------------------------------------------------------------------

#define IGNORE_INDEX 255

enum : int {
  C_    = 19,
  B_    = 8,
  H_    = 512,
  W_    = 1024,
  HW_   = H_ * W_,          // 524288 (2^19)
  N_    = B_ * HW_,         // 4194304 pixels
  TILE_ = 1024,             // one (b,h) row per tile
  NT_   = N_ / TILE_,       // 4096 tiles
  WGS_  = 256,              // 8 waves (wave32)
  NWG_  = 512,              // workgroups for streaming passes
  TPW_  = NT_ / NWG_,       // 8 tiles per workgroup in pass1
  NB1_  = 2048,             // bits[31:21]
  NB2_  = 2048,             // bits[20:10]
  NB3_  = 1024,             // bits[9:0]
  NSCAL_ = 16,
};

// ws layout (bytes)
#define LOSS_OFF  0
#define HIST1_OFF ((size_t)N_ * 4)
#define HIST2_OFF (HIST1_OFF + (size_t)NB1_ * 4)
#define HIST3_OFF (HIST2_OFF + (size_t)NB2_ * 4)
#define SCAL_OFF  (HIST3_OFF + (size_t)NB3_ * 4)
#define PART_OFF  (SCAL_OFF + (size_t)NSCAL_ * 4)   // 8-byte aligned
#define CNTS_OFF  (PART_OFF + (size_t)NWG_ * 8)

// scal[] indices: 0=n_valid 1=n_keep 2=use_thresh 3=b1 4=k1 5=b2 6=k2 7=thr_bits

// --- CDNA5 async global->LDS staging ---------------------------------------
// Leading s_wait_dscnt 0 guarantees the previous ds_load of the target buffer
// has landed in VGPRs before the async engine overwrites it. The "memory"
// clobber orders this against surrounding C++ LDS accesses.
#define ASYNC_LOAD_B128(ldsaddr, gofs, sbase)                                  \
  asm volatile("s_wait_dscnt 0x0\n\t"                                          \
               "global_load_async_to_lds_b128 %0, %1, %2"                      \
               :: "v"(ldsaddr), "v"(gofs), "s"(sbase) : "memory")

#define WAIT_ASYNC_1() asm volatile("s_wait_asynccnt 0x1" ::: "memory")
#define WAIT_ASYNC_0() asm volatile("s_wait_asynccnt 0x0" ::: "memory")

__device__ __forceinline__ void lse_update(float v, int c, int lab,
                                           float& m, float& s, float& xl) {
  float nm = fmaxf(m, v);
  s = s * __expf(m - nm) + __expf(v - nm);
  m = nm;
  if (c == lab) xl = v;
}

__global__ void k_init(unsigned* __restrict__ p) {
  for (int i = (int)threadIdx.x; i < NB1_ + NB2_ + NB3_ + NSCAL_; i += (int)blockDim.x)
    p[i] = 0u;
}

__global__ __launch_bounds__(WGS_) void k_pass1(
    const float* __restrict__ logits, const int* __restrict__ labels,
    float* __restrict__ loss, unsigned* __restrict__ hist1,
    unsigned* __restrict__ scal) {
  __shared__ float4 stage[8 * 2 * 32];   // [wave][buf][lane], 8 KB
  __shared__ unsigned lhist[NB1_];       // 8 KB
  __shared__ unsigned lcnt;

  const int t = (int)threadIdx.x;
  const int wave = t >> 5, lane = t & 31;
  for (int i = t; i < NB1_; i += WGS_) lhist[i] = 0u;
  if (t == 0) lcnt = 0u;
  __syncthreads();

  const uint64_t base = (uint64_t)(uintptr_t)logits;
  // low 32 bits of a generic LDS pointer == wave-relative LDS byte offset
  const unsigned lds0 = (unsigned)(uintptr_t)(&stage[(wave * 2 + 0) * 32 + lane]);
  const unsigned lds1 = lds0 + 512u;     // second buffer: +32 float4
  unsigned nval = 0u;

  for (int it = 0; it < TPW_; ++it) {
    const int tile = (int)blockIdx.x * TPW_ + it;
    const int p0 = tile * TILE_;
    const int bb = p0 >> 19;                 // / HW_
    const int hw = p0 & (HW_ - 1);
    // byte offset of this thread's 16B piece of class c's 4KB row chunk
    const unsigned g0 = 4u * ((unsigned)bb * (unsigned)(C_ * HW_) + (unsigned)hw)
                        + 16u * (unsigned)t;

    ASYNC_LOAD_B128(lds0, g0, base);                       // class 0
    ASYNC_LOAD_B128(lds1, g0 + 4u * (unsigned)HW_, base);  // class 1

    const int4 lab = *(const int4*)(labels + p0 + 4 * t);

    float4 m, s, xl;
    m.x = m.y = m.z = m.w = -__builtin_inff();
    s.x = s.y = s.z = s.w = 0.0f;
    xl.x = xl.y = xl.z = xl.w = 0.0f;

#pragma unroll
    for (int c = 0; c < C_; ++c) {
      if (c < C_ - 1) { WAIT_ASYNC_1(); } else { WAIT_ASYNC_0(); }
      float4 v = stage[(wave * 2 + (c & 1)) * 32 + lane];  // ds_load_b128
      if (c + 2 < C_) {
        ASYNC_LOAD_B128((c & 1) ? lds1 : lds0,
                        g0 + 4u * (unsigned)HW_ * (unsigned)(c + 2), base);
      }
      lse_update(v.x, c, lab.x, m.x, s.x, xl.x);
      lse_update(v.y, c, lab.y, m.y, s.y, xl.y);
      lse_update(v.z, c, lab.z, m.z, s.z, xl.z);
      lse_update(v.w, c, lab.w, m.w, s.w, xl.w);
    }

    unsigned bits[4];
    bits[0] = (lab.x == IGNORE_INDEX) ? 0xFFFFFFFFu
                                      : __float_as_uint((m.x + __logf(s.x)) - xl.x);
    bits[1] = (lab.y == IGNORE_INDEX) ? 0xFFFFFFFFu
                                      : __float_as_uint((m.y + __logf(s.y)) - xl.y);
    bits[2] = (lab.z == IGNORE_INDEX) ? 0xFFFFFFFFu
                                      : __float_as_uint((m.z + __logf(s.z)) - xl.z);
    bits[3] = (lab.w == IGNORE_INDEX) ? 0xFFFFFFFFu
                                      : __float_as_uint((m.w + __logf(s.w)) - xl.w);
#pragma unroll
    for (int j = 0; j < 4; ++j) {
      if ((int)bits[j] >= 0) {   // valid nll is >= +0.0f; sentinel has sign bit
        ++nval;
        atomicAdd(&lhist[bits[j] >> 21], 1u);
      }
    }
    float4 out;
    out.x = __uint_as_float(bits[0]);
    out.y = __uint_as_float(bits[1]);
    out.z = __uint_as_float(bits[2]);
    out.w = __uint_as_float(bits[3]);
    *(float4*)(loss + p0 + 4 * t) = out;
  }

  atomicAdd(&lcnt, nval);
  __syncthreads();
  if (t == 0) atomicAdd(&scal[0], lcnt);
  for (int i = t; i < NB1_; i += WGS_) {
    unsigned h = lhist[i];
    if (h) atomicAdd(&hist1[i], h);
  }
}

__global__ void k_scan(unsigned* __restrict__ scal,
                       const unsigned* __restrict__ hist, int level) {
  if (threadIdx.x != 0 || blockIdx.x != 0) return;
  if (level == 0) {
    int nv = (int)scal[0];
    int nk = (int)(0.3f * (float)nv);            // matches fp32 mul + int cast
    int mk = nv < 100000 ? nv : 100000;
    if (nk < mk) nk = mk;
    if (nk > nv) nk = nv;
    scal[1] = (unsigned)nk;
    unsigned use = (nk < nv) ? 1u : 0u;
    scal[2] = use;
    if (use) {
      unsigned k = (unsigned)nk, cum = 0u;
      for (int i = NB1_ - 1; i >= 0; --i) {
        unsigned h = hist[i];
        if (cum + h >= k) { scal[3] = (unsigned)i; scal[4] = k - cum; break; }
        cum += h;
      }
    }
  } else if (level == 1) {
    if (!scal[2]) return;
    unsigned k = scal[4], cum = 0u;
    for (int i = NB2_ - 1; i >= 0; --i) {
      unsigned h = hist[i];
      if (cum + h >= k) { scal[5] = (unsigned)i; scal[6] = k - cum; break; }
      cum += h;
    }
  } else {
    if (!scal[2]) { scal[7] = 0u; return; }
    unsigned k = scal[6], cum = 0u;
    for (int i = NB3_ - 1; i >= 0; --i) {
      unsigned h = hist[i];
      if (cum + h >= k) {
        scal[7] = (scal[3] << 21) | (scal[5] << 10) | (unsigned)i;
        break;
      }
      cum += h;
    }
  }
}

__global__ __launch_bounds__(WGS_) void k_histmid(
    const float* __restrict__ loss, unsigned* __restrict__ histo,
    const unsigned* __restrict__ scal, int level) {
  __shared__ unsigned lh[NB1_];
  const int t = (int)threadIdx.x;
  const int nb = (level == 1) ? NB2_ : NB3_;
  for (int i = t; i < nb; i += WGS_) lh[i] = 0u;
  __syncthreads();
  if (scal[2]) {
    unsigned key, shift, sub, mask;
    if (level == 1) { key = scal[3]; shift = 21u; sub = 10u; mask = 2047u; }
    else { key = (scal[3] << 11) | scal[5]; shift = 10u; sub = 0u; mask = 1023u; }
    const int stride = (int)gridDim.x * WGS_;
    for (int i = (int)blockIdx.x * WGS_ + t; i < N_ / 4; i += stride) {
      float4 v = ((const float4*)loss)[i];
      unsigned bb[4] = {__float_as_uint(v.x), __float_as_uint(v.y),
                        __float_as_uint(v.z), __float_as_uint(v.w)};
#pragma unroll
      for (int j = 0; j < 4; ++j)
        if ((int)bb[j] >= 0 && (bb[j] >> shift) == key)
          atomicAdd(&lh[(bb[j] >> sub) & mask], 1u);
    }
  }
  __syncthreads();
  for (int i = t; i < nb; i += WGS_) {
    unsigned h = lh[i];
    if (h) atomicAdd(&histo[i], h);
  }
}

__global__ __launch_bounds__(WGS_) void k_partial(
    const float* __restrict__ loss, const unsigned* __restrict__ scal,
    double* __restrict__ part, unsigned* __restrict__ cnts) {
  const unsigned use = scal[2];
  const float thr = __uint_as_float(scal[7]);
  double sum = 0.0;
  unsigned cnt = 0u;
  const int t = (int)threadIdx.x;
  const int stride = (int)gridDim.x * WGS_;
  for (int i = (int)blockIdx.x * WGS_ + t; i < N_ / 4; i += stride) {
    float4 v = ((const float4*)loss)[i];
    float a[4] = {v.x, v.y, v.z, v.w};
#pragma unroll
    for (int j = 0; j < 4; ++j) {
      unsigned b = __float_as_uint(a[j]);
      if ((int)b >= 0 && (!use || a[j] >= thr)) { sum += (double)a[j]; ++cnt; }
    }
  }
  __shared__ double sd[WGS_];
  __shared__ unsigned sc[WGS_];
  sd[t] = sum; sc[t] = cnt;
  __syncthreads();
  for (int o = WGS_ / 2; o > 0; o >>= 1) {
    if (t < o) { sd[t] += sd[t + o]; sc[t] += sc[t + o]; }
    __syncthreads();
  }
  if (t == 0) { part[blockIdx.x] = sd[0]; cnts[blockIdx.x] = sc[0]; }
}

__global__ __launch_bounds__(WGS_) void k_final(
    const double* __restrict__ part, const unsigned* __restrict__ cnts,
    const unsigned* __restrict__ scal, float* __restrict__ out) {
  __shared__ double sd[WGS_];
  __shared__ unsigned sc[WGS_];
  const int t = (int)threadIdx.x;
  sd[t] = part[t] + part[t + WGS_];
  sc[t] = cnts[t] + cnts[t + WGS_];
  __syncthreads();
  for (int o = WGS_ / 2; o > 0; o >>= 1) {
    if (t < o) { sd[t] += sd[t + o]; sc[t] += sc[t + o]; }
    __syncthreads();
  }
  if (t == 0) {
    float r = 0.0f;
    if ((int)scal[0] > 0) {
      double cc = (double)(sc[0] ? sc[0] : 1u);
      r = (float)(sd[0] / cc);
    }
    out[0] = r;
  }
}

extern "C" void kernel_launch(void* const* d_in, const int* in_sizes, int n_in,
                              void* d_out, int out_size, void* d_ws, size_t ws_size,
                              hipStream_t stream) {
  (void)in_sizes; (void)n_in; (void)out_size; (void)ws_size;
  const float* logits = (const float*)d_in[0];
  const int* labels = (const int*)d_in[1];
  float* out = (float*)d_out;
  char* ws = (char*)d_ws;

  float* loss     = (float*)(ws + LOSS_OFF);
  unsigned* hist1 = (unsigned*)(ws + HIST1_OFF);
  unsigned* hist2 = (unsigned*)(ws + HIST2_OFF);
  unsigned* hist3 = (unsigned*)(ws + HIST3_OFF);
  unsigned* scal  = (unsigned*)(ws + SCAL_OFF);
  double* part    = (double*)(ws + PART_OFF);
  unsigned* cnts  = (unsigned*)(ws + CNTS_OFF);

  k_init<<<1, WGS_, 0, stream>>>(hist1);  // hist1/hist2/hist3/scal contiguous
  k_pass1<<<NWG_, WGS_, 0, stream>>>(logits, labels, loss, hist1, scal);
  k_scan<<<1, 1, 0, stream>>>(scal, hist1, 0);
  k_histmid<<<NWG_, WGS_, 0, stream>>>(loss, hist2, scal, 1);
  k_scan<<<1, 1, 0, stream>>>(scal, hist2, 1);
  k_histmid<<<NWG_, WGS_, 0, stream>>>(loss, hist3, scal, 2);
  k_scan<<<1, 1, 0, stream>>>(scal, hist3, 2);
  k_partial<<<NWG_, WGS_, 0, stream>>>(loss, scal, part, cnts);
  k_final<<<1, WGS_, 0, stream>>>(part, cnts, scal, out);
}